// ConvOnly_25305947308672
// MI455X (gfx1250) — compile-verified
//
#include <hip/hip_runtime.h>

#define NTRIP 16384
#define NREL  237
#define DDIM  128
#define NCH   50
#define MAX_TILES (NREL + NTRIP / 16)   // 1261

typedef __attribute__((ext_vector_type(2))) float v2f;
typedef __attribute__((ext_vector_type(8))) float v8f;

// ---------------------------------------------------------------- utilities
__global__ void k_zero(int* __restrict__ p, int n) {
    int i = blockIdx.x * blockDim.x + threadIdx.x;
    if (i < n) p[i] = 0;
}

__global__ void k_count(const int* __restrict__ triples, int* __restrict__ counts) {
    int b = blockIdx.x * blockDim.x + threadIdx.x;
    if (b < NTRIP) atomicAdd(&counts[triples[3 * b + 1]], 1);
}

// Single-thread serial tile build: 237 relations + <=1261 tile records. Trivial cost.
__global__ void k_build(const int* __restrict__ counts, int* __restrict__ offsets,
                        int* __restrict__ tile_rel, int* __restrict__ tile_start,
                        int* __restrict__ tile_rows, int* __restrict__ num_tiles) {
    if (threadIdx.x == 0 && blockIdx.x == 0) {
        int off = 0, t = 0;
        for (int r = 0; r < NREL; ++r) {
            offsets[r] = off;
            int c = counts[r];
            for (int s = 0; s < c; s += 16) {
                tile_rel[t]   = r;
                tile_start[t] = off + s;
                int rem = c - s;
                tile_rows[t]  = rem < 16 ? rem : 16;
                ++t;
            }
            off += c;
        }
        num_tiles[0] = t;
    }
}

__global__ void k_scatter(const int* __restrict__ triples, const int* __restrict__ offsets,
                          int* __restrict__ cursors, int* __restrict__ perm) {
    int b = blockIdx.x * blockDim.x + threadIdx.x;
    if (b < NTRIP) {
        int r   = triples[3 * b + 1];
        int pos = offsets[r] + atomicAdd(&cursors[r], 1);
        perm[pos] = b;
    }
}

// ---------------------------------------------------------------- main kernel
static __device__ inline v8f wmma_f32_k4(v2f a, v2f b, v8f c) {
    // D = A(16x4,f32) * B(4x16,f32) + C(16x16,f32)
    return __builtin_amdgcn_wmma_f32_16x16x4_f32(
        /*neg_a=*/false, a, /*neg_b=*/false, b,
        /*c_mod=*/(short)0, c, /*reuse_a=*/false, /*reuse_b=*/false);
}

__global__ __launch_bounds__(128) void k_main(
    const int*   __restrict__ triples,
    const float* __restrict__ ent_emb,
    const float* __restrict__ rel_emb,
    const float* __restrict__ rel_W,
    const float* __restrict__ conv_w,
    const float* __restrict__ conv_b,
    const float* __restrict__ fc_w,
    const float* __restrict__ fc_b,
    const int*   __restrict__ tile_rel,
    const int*   __restrict__ tile_start,
    const int*   __restrict__ tile_rows,
    const int*   __restrict__ num_tiles,
    const int*   __restrict__ perm,
    float*       __restrict__ out)
{
    __shared__ __align__(16) float s_in [2][16][DDIM];  // gathered head/tail rows
    __shared__ __align__(16) float s_out[2][16][DDIM];  // ph / pt results
    __shared__ __align__(16) float s_rel[DDIM];
    __shared__ int s_bidx[16], s_hidx[16], s_tidx[16];
    __shared__ float s_red[16][8];
    __shared__ int s_nt;

    if (threadIdx.x == 0) s_nt = num_tiles[0];
    __syncthreads();
    int t = blockIdx.x;
    if (t >= s_nt) return;

    int rel   = tile_rel[t];
    int start = tile_start[t];
    int rows  = tile_rows[t];

    if (threadIdx.x < 16) {
        int m = threadIdx.x;
        int b = (m < rows) ? perm[start + m] : -1;
        s_bidx[m] = b;
        s_hidx[m] = (b >= 0) ? triples[3 * b + 0] : 0;
        s_tidx[m] = (b >= 0) ? triples[3 * b + 2] : 0;
    }
    __syncthreads();

    // Cooperative gather of 32 embedding rows (16 head + 16 tail), float4 vectorized.
    for (int i = threadIdx.x; i < 32 * 32; i += 128) {
        int row = i >> 5, q = i & 31;
        int side = row >> 4, rm = row & 15;
        int idx = side ? s_tidx[rm] : s_hidx[rm];
        ((float4*)&s_in[side][rm][0])[q] =
            ((const float4*)(ent_emb + (size_t)idx * DDIM))[q];
    }
    if (threadIdx.x < 32)
        ((float4*)s_rel)[threadIdx.x] =
            ((const float4*)(rel_emb + (size_t)rel * DDIM))[threadIdx.x];
    __syncthreads();

    // ---- WMMA phase: [16 x 128] = [16 x 128] x [128 x 128], f32, K-step 4.
    // Wave w owns output columns [32w, 32w+32): two 16-wide column tiles.
    // B fragments of W are shared between the ph and pt accumulations.
    int wv   = threadIdx.x >> 5;
    int lane = threadIdx.x & 31;
    int half = lane >> 4;       // lane group (selects K sub-pair / M upper half)
    int l15  = lane & 15;
    const float* Wr = rel_W + (size_t)rel * DDIM * DDIM;

    v8f acc_ph0 = {}, acc_ph1 = {}, acc_pt0 = {}, acc_pt1 = {};
    int e0 = (2 * wv) * 16 + l15;   // column for tile 0
    int e1 = e0 + 16;               // column for tile 1

#pragma unroll 4
    for (int kk = 0; kk < DDIM; kk += 4) {
        int ka = kk + half * 2;     // A VGPR0/1 hold K = ka, ka+1 (per 16x4 f32 layout)
        v2f a_ph, a_pt, b0, b1;
        a_ph.x = s_in[0][l15][ka];     a_ph.y = s_in[0][l15][ka + 1];
        a_pt.x = s_in[1][l15][ka];     a_pt.y = s_in[1][l15][ka + 1];
        b0.x = Wr[(size_t)ka * DDIM + e0];  b0.y = Wr[(size_t)(ka + 1) * DDIM + e0];
        b1.x = Wr[(size_t)ka * DDIM + e1];  b1.y = Wr[(size_t)(ka + 1) * DDIM + e1];
        acc_ph0 = wmma_f32_k4(a_ph, b0, acc_ph0);
        acc_ph1 = wmma_f32_k4(a_ph, b1, acc_ph1);
        acc_pt0 = wmma_f32_k4(a_pt, b0, acc_pt0);
        acc_pt1 = wmma_f32_k4(a_pt, b1, acc_pt1);
    }

    // C/D layout: VGPR i -> (M = i + 8*half, N = l15)
#pragma unroll
    for (int i = 0; i < 8; ++i) {
        int M = i + 8 * half;
        s_out[0][M][e0] = acc_ph0[i];
        s_out[0][M][e1] = acc_ph1[i];
        s_out[1][M][e0] = acc_pt0[i];
        s_out[1][M][e1] = acc_pt1[i];
    }
    __syncthreads();

    // ---- conv (3-tap over {ph, rel, pt}) + ReLU + fc dot, fully in VALU.
    // thread -> (triple m = tid/8, 16-wide d-segment = tid%8)
    int m   = threadIdx.x >> 3;
    int d0  = (threadIdx.x & 7) * 16;
    float partial = 0.f;
#pragma unroll 1
    for (int c = 0; c < NCH; ++c) {
        float w0 = conv_w[3 * c + 0];
        float w1 = conv_w[3 * c + 1];
        float w2 = conv_w[3 * c + 2];
        float bc = conv_b[c];
        const float* fw = fc_w + c * DDIM + d0;
#pragma unroll
        for (int d = 0; d < 16; ++d) {
            float v = fmaf(w0, s_out[0][m][d0 + d],
                      fmaf(w1, s_rel[d0 + d],
                      fmaf(w2, s_out[1][m][d0 + d], bc)));
            v = v > 0.f ? v : 0.f;
            partial = fmaf(v, fw[d], partial);
        }
    }
    s_red[m][threadIdx.x & 7] = partial;
    __syncthreads();

    if (threadIdx.x < 16) {
        float s = 0.f;
#pragma unroll
        for (int i = 0; i < 8; ++i) s += s_red[threadIdx.x][i];
        if (threadIdx.x < rows) {
            int b = s_bidx[threadIdx.x];
            out[b] = s + fc_b[0];
        }
    }
}

// ---------------------------------------------------------------- launch
extern "C" void kernel_launch(void* const* d_in, const int* in_sizes, int n_in,
                              void* d_out, int out_size, void* d_ws, size_t ws_size,
                              hipStream_t stream) {
    const int*   triples = (const int*)  d_in[0];
    const float* ent_emb = (const float*)d_in[1];
    const float* rel_emb = (const float*)d_in[2];
    const float* rel_W   = (const float*)d_in[3];
    const float* conv_w  = (const float*)d_in[4];
    const float* conv_b  = (const float*)d_in[5];
    const float* fc_w    = (const float*)d_in[6];
    const float* fc_b    = (const float*)d_in[7];
    float* out = (float*)d_out;

    int* wsi        = (int*)d_ws;
    int* counts     = wsi;                      // 256
    int* offsets    = wsi + 256;                // 256
    int* cursors    = wsi + 512;                // 256
    int* num_tiles  = wsi + 768;                // 4 (padded)
    int* tile_rel   = wsi + 772;                // MAX_TILES
    int* tile_start = tile_rel + MAX_TILES;
    int* tile_rows  = tile_start + MAX_TILES;
    int* perm       = tile_rows + MAX_TILES;    // NTRIP

    k_zero   <<<3, 256, 0, stream>>>(wsi, 768);            // counts + cursors (offsets rebuilt)
    k_count  <<<NTRIP / 256, 256, 0, stream>>>(triples, counts);
    k_build  <<<1, 32, 0, stream>>>(counts, offsets, tile_rel, tile_start, tile_rows, num_tiles);
    k_scatter<<<NTRIP / 256, 256, 0, stream>>>(triples, offsets, cursors, perm);
    k_main   <<<MAX_TILES, 128, 0, stream>>>(triples, ent_emb, rel_emb, rel_W,
                                             conv_w, conv_b, fc_w, fc_b,
                                             tile_rel, tile_start, tile_rows, num_tiles,
                                             perm, out);
}